// GraphConvolutionalLayer_31310311588446
// MI455X (gfx1250) — compile-verified
//
#include <hip/hip_runtime.h>

// ---------------------------------------------------------------------------
// RGCN layer (mean aggregation per (dst, relation) + per-relation linear +
// root linear + bias + ReLU) for MI455X / gfx1250.
//
// Fast path:
//   1) cnt[dst*16+rel] += 1                         (u32 atomics, 3.2 MB, L2)
//   2) agg[dst,rel,:]  += x[src,:] / cnt            (f32 atomics; mean folded in)
//   3) repack [W_rel ; W_root] into WMMA lane order (one b64 per B fragment)
//   4) out = relu( [agg | x] @ Bpack + bias )       via V_WMMA_F32_16X16X4_F32,
//      2 M-tiles per wave so each B fragment feeds two WMMAs.
//
// Fallback (workspace too small): transform-then-aggregate per edge.
// ---------------------------------------------------------------------------

constexpr int kNodes = 50000;
constexpr int kEdges = 1600000;
constexpr int kDIn   = 128;
constexpr int kDOut  = 128;
constexpr int kRel   = 16;
constexpr int kKAgg  = kRel * kDIn;          // 2048 flattened (rel, d_in)
constexpr int kTAgg  = kKAgg / 4;            // 512 K-steps over agg
constexpr int kTRoot = kDIn / 4;             // 32 K-steps over x/W_root
constexpr int kTTot  = kTAgg + kTRoot;       // 544 packed K-steps
constexpr int kMTiles = kNodes / 16;         // 3125 (exact)
constexpr int kPairs  = (kMTiles + 1) / 2;   // 1563 wave work items

typedef __attribute__((ext_vector_type(2))) float v2f;
typedef __attribute__((ext_vector_type(8))) float v8f;

// ---------------------------------------------------------------------------
// Pass 1: per-(dst, rel) edge counts.
// ---------------------------------------------------------------------------
__global__ __launch_bounds__(256) void rgcn_count(const int* __restrict__ dst,
                                                  const int* __restrict__ etype,
                                                  unsigned* __restrict__ cnt) {
  const int e = blockIdx.x * blockDim.x + threadIdx.x;
  if (e < kEdges) {
    atomicAdd(&cnt[dst[e] * kRel + etype[e]], 1u);
  }
}

// ---------------------------------------------------------------------------
// Pass 2: scatter x[src]/cnt into agg[(dst*16+rel), 128].  One wave per edge;
// each lane moves a float4 (32 lanes * 4 = 128 features) -> coalesced 512B
// gather + native global_atomic_add_f32 scatter.
// ---------------------------------------------------------------------------
__global__ __launch_bounds__(256) void rgcn_scatter(const float* __restrict__ x,
                                                    const int* __restrict__ src,
                                                    const int* __restrict__ dst,
                                                    const int* __restrict__ etype,
                                                    const unsigned* __restrict__ cnt,
                                                    float* __restrict__ agg) {
  const int e    = (int)((blockIdx.x * blockDim.x + threadIdx.x) >> 5);
  const int lane = threadIdx.x & 31;
  if (e >= kEdges) return;
  const int s   = src[e];
  const int seg = dst[e] * kRel + etype[e];
  const unsigned c = cnt[seg];
  const float inv = 1.0f / (float)(c ? c : 1u);
  const float4 v = ((const float4*)(x + (size_t)s * kDIn))[lane];
  float* a = agg + (size_t)seg * kDIn + lane * 4;
  unsafeAtomicAdd(a + 0, v.x * inv);
  unsafeAtomicAdd(a + 1, v.y * inv);
  unsafeAtomicAdd(a + 2, v.z * inv);
  unsafeAtomicAdd(a + 3, v.w * inv);
}

// ---------------------------------------------------------------------------
// Pass 3a: repack [W_rel(2048x128) ; W_root(128x128)] into per-lane WMMA
// B-fragment order:
//   Bpack[((t*8 + n)*32 + lane)] = { B[kk][c], B[kk+1][c] }   (float2)
//   kk = 4t + 2*(lane>>4), c = 16n + (lane&15)
// so the GEMM reads one coalesced global_load_b64 per (t, n) fragment.
// ---------------------------------------------------------------------------
__global__ __launch_bounds__(256) void rgcn_pack_b(const float* __restrict__ wrel,
                                                   const float* __restrict__ wroot,
                                                   float* __restrict__ bpack) {
  const int idx = blockIdx.x * blockDim.x + threadIdx.x;  // (t*8+n)*32 + lane
  if (idx >= kTTot * 8 * 32) return;
  const int lane = idx & 31;
  const int n    = (idx >> 5) & 7;
  const int t    = idx >> 8;
  const int lm = lane & 15;
  const int lk = lane >> 4;
  const int c  = n * 16 + lm;
  const float* w;
  int kk;
  if (t < kTAgg) { w = wrel;  kk = 4 * t + 2 * lk; }
  else           { w = wroot; kk = 4 * (t - kTAgg) + 2 * lk; }
  float2 v;
  v.x = w[(size_t)kk * kDOut + c];
  v.y = w[(size_t)(kk + 1) * kDOut + c];
  ((float2*)bpack)[idx] = v;
}

// ---------------------------------------------------------------------------
// Pass 3b: fused GEMM  out[N,128] = relu([agg|x][N,2176] @ B[2176,128] + bias)
// with V_WMMA_F32_16X16X4_F32 (full fp32 precision; matmul phase stays near
// the HBM roofline for the 435 MB read-once agg stream).
//
// One wave owns TWO 16x128 output tiles (rows row0.. and row1..):
//   per K-step: 2 NT b64 A loads + 8 b64 packed-B loads -> 16 WMMAs.
// 3125 tiles is odd, so the tail wave clamps tile1 onto tile0 (recomputes and
// rewrites identical values; no cross-wave aliasing).
// ---------------------------------------------------------------------------
__global__ __launch_bounds__(256) void rgcn_gemm(const float* __restrict__ agg,
                                                 const float* __restrict__ x,
                                                 const float* __restrict__ bpack,
                                                 const float* __restrict__ bias,
                                                 float* __restrict__ out) {
  const int lane = threadIdx.x & 31;
  const int wave = threadIdx.x >> 5;
  const int pair = blockIdx.x * 8 + wave;
  if (pair >= kPairs) return;   // wave-uniform; EXEC stays all-ones for WMMA

  int row0 = pair * 32;
  int row1 = row0 + 16;
  if (row1 + 16 > kNodes) row1 = row0;   // odd tail: duplicate tile

  const int lm = lane & 15;   // M index (A) / N index (B, C/D)
  const int lk = lane >> 4;   // selects K pair {0,1} vs {2,3}

  v8f acc0[8], acc1[8];
  const v8f vzero = {0.f, 0.f, 0.f, 0.f, 0.f, 0.f, 0.f, 0.f};
#pragma unroll
  for (int n = 0; n < 8; ++n) { acc0[n] = vzero; acc1[n] = vzero; }

  const float* bl = bpack + lane * 2;   // + t*512 + n*64

  // --- Phase 1: K over flattened (rel, d_in); A = agg (read-once -> NT)
  {
    const float* arow0 = agg + (size_t)(row0 + lm) * kKAgg + 2 * lk;
    const float* arow1 = agg + (size_t)(row1 + lm) * kKAgg + 2 * lk;
    for (int t = 0; t < kTAgg; ++t) {
      const v2f a0 = __builtin_nontemporal_load((const v2f*)(arow0 + 4 * t));
      const v2f a1 = __builtin_nontemporal_load((const v2f*)(arow1 + 4 * t));
      const float* bt = bl + (size_t)t * 512;
#pragma unroll
      for (int n = 0; n < 8; ++n) {
        const v2f b = *(const v2f*)(bt + n * 64);
        acc0[n] = __builtin_amdgcn_wmma_f32_16x16x4_f32(
            false, a0, false, b, (short)0, acc0[n], false, false);
        acc1[n] = __builtin_amdgcn_wmma_f32_16x16x4_f32(
            false, a1, false, b, (short)0, acc1[n], false, false);
      }
    }
  }

  // --- Phase 2: K over d_in; A = x, B = packed W_root tail
  {
    const float* xrow0 = x + (size_t)(row0 + lm) * kDIn + 2 * lk;
    const float* xrow1 = x + (size_t)(row1 + lm) * kDIn + 2 * lk;
    for (int t = 0; t < kTRoot; ++t) {
      const v2f a0 = *(const v2f*)(xrow0 + 4 * t);
      const v2f a1 = *(const v2f*)(xrow1 + 4 * t);
      const float* bt = bl + (size_t)(kTAgg + t) * 512;
#pragma unroll
      for (int n = 0; n < 8; ++n) {
        const v2f b = *(const v2f*)(bt + n * 64);
        acc0[n] = __builtin_amdgcn_wmma_f32_16x16x4_f32(
            false, a0, false, b, (short)0, acc0[n], false, false);
        acc1[n] = __builtin_amdgcn_wmma_f32_16x16x4_f32(
            false, a1, false, b, (short)0, acc1[n], false, false);
      }
    }
  }

  // --- Epilogue: + bias, ReLU, streaming stores (both tiles)
#pragma unroll
  for (int n = 0; n < 8; ++n) {
    const int col = n * 16 + lm;
    const float b = bias[col];
#pragma unroll
    for (int v = 0; v < 8; ++v) {
      const int r0 = row0 + v + lk * 8;
      const int r1 = row1 + v + lk * 8;
      float v0 = acc0[n][v] + b;
      float v1 = acc1[n][v] + b;
      v0 = v0 > 0.f ? v0 : 0.f;
      v1 = v1 > 0.f ? v1 : 0.f;
      __builtin_nontemporal_store(v0, out + (size_t)r0 * kDOut + col);
      __builtin_nontemporal_store(v1, out + (size_t)r1 * kDOut + col);
    }
  }
}

// ---------------------------------------------------------------------------
// Fallback path (workspace too small for the ~412 MB agg buffer):
// transform-then-aggregate per edge.
// ---------------------------------------------------------------------------
__global__ __launch_bounds__(256) void fb_root_bias(const float* __restrict__ x,
                                                    const float* __restrict__ wroot,
                                                    const float* __restrict__ bias,
                                                    float* __restrict__ out) {
  const int idx = blockIdx.x * blockDim.x + threadIdx.x;
  if (idx >= kNodes * kDOut) return;
  const int n = idx / kDOut, o = idx % kDOut;
  const float* xr = x + (size_t)n * kDIn;
  float s = bias[o];
  for (int i = 0; i < kDIn; ++i) s += xr[i] * wroot[i * kDOut + o];
  out[idx] = s;
}

__global__ __launch_bounds__(256) void fb_edge_apply(const float* __restrict__ x,
                                                     const int* __restrict__ src,
                                                     const int* __restrict__ dst,
                                                     const int* __restrict__ etype,
                                                     const unsigned* __restrict__ cnt,
                                                     const float* __restrict__ wrel,
                                                     float* __restrict__ out) {
  const int e    = (int)((blockIdx.x * blockDim.x + threadIdx.x) >> 5);
  const int lane = threadIdx.x & 31;
  if (e >= kEdges) return;
  const int s = src[e], d = dst[e], r = etype[e];
  const unsigned c = cnt[d * kRel + r];
  const float inv = 1.0f / (float)(c ? c : 1u);
  const float* xs = x + (size_t)s * kDIn;
  const float* w  = wrel + (size_t)r * kDIn * kDOut + lane * 4;
  float a0 = 0.f, a1 = 0.f, a2 = 0.f, a3 = 0.f;
  for (int i = 0; i < kDIn; ++i) {
    const float xv = xs[i];
    const float* wp = w + (size_t)i * kDOut;
    a0 += xv * wp[0]; a1 += xv * wp[1]; a2 += xv * wp[2]; a3 += xv * wp[3];
  }
  float* o = out + (size_t)d * kDOut + lane * 4;
  unsafeAtomicAdd(o + 0, a0 * inv);
  unsafeAtomicAdd(o + 1, a1 * inv);
  unsafeAtomicAdd(o + 2, a2 * inv);
  unsafeAtomicAdd(o + 3, a3 * inv);
}

__global__ __launch_bounds__(256) void fb_relu(float* __restrict__ out) {
  const int i = blockIdx.x * blockDim.x + threadIdx.x;
  if (i < kNodes * kDOut) {
    const float v = out[i];
    out[i] = v > 0.f ? v : 0.f;
  }
}

// ---------------------------------------------------------------------------
extern "C" void kernel_launch(void* const* d_in, const int* in_sizes, int n_in,
                              void* d_out, int out_size, void* d_ws, size_t ws_size,
                              hipStream_t stream) {
  (void)in_sizes; (void)n_in; (void)out_size;
  // setup_inputs() order:
  //   0: node_features f32 [N,128]   1: edge_index int [2,E]
  //   2: edge_type int [E]           3: W_rel f32 [16,128,128]
  //   4: W_root f32 [128,128]        5: bias f32 [128]
  const float* x     = (const float*)d_in[0];
  const int*   eidx  = (const int*)d_in[1];
  const int*   etype = (const int*)d_in[2];
  const float* wrel  = (const float*)d_in[3];   // [16,128,128] == row-major [2048,128]
  const float* wroot = (const float*)d_in[4];
  const float* bias  = (const float*)d_in[5];
  float* out = (float*)d_out;

  const int* src = eidx;            // edge_index[0, :]
  const int* dst = eidx + kEdges;   // edge_index[1, :]

  // Workspace carve-up: [cnt u32][agg f32][Bpack f32]
  unsigned* cnt = (unsigned*)d_ws;
  const size_t cnt_bytes = (size_t)kNodes * kRel * sizeof(unsigned);
  const size_t cnt_pad   = (cnt_bytes + 255) & ~(size_t)255;
  float* agg = (float*)((char*)d_ws + cnt_pad);
  const size_t agg_bytes = (size_t)kNodes * kRel * kDIn * sizeof(float);
  float* bpack = (float*)((char*)d_ws + cnt_pad + agg_bytes);
  const size_t bpack_bytes = (size_t)kTTot * 8 * 32 * 2 * sizeof(float);

  hipMemsetAsync(cnt, 0, cnt_bytes, stream);
  rgcn_count<<<(kEdges + 255) / 256, 256, 0, stream>>>(dst, etype, cnt);

  if (ws_size >= cnt_pad + agg_bytes + bpack_bytes) {
    // Fast path: aggregate-then-transform with packed-B WMMA GEMM.
    hipMemsetAsync(agg, 0, agg_bytes, stream);
    rgcn_pack_b<<<(kTTot * 8 * 32 + 255) / 256, 256, 0, stream>>>(wrel, wroot, bpack);
    rgcn_scatter<<<(kEdges * 32) / 256, 256, 0, stream>>>(x, src, dst, etype, cnt, agg);
    rgcn_gemm<<<(kPairs + 7) / 8, 256, 0, stream>>>(agg, x, bpack, bias, out);
  } else {
    // Fallback: transform-then-aggregate per edge.
    fb_root_bias<<<(kNodes * kDOut + 255) / 256, 256, 0, stream>>>(x, wroot, bias, out);
    fb_edge_apply<<<(kEdges * 32) / 256, 256, 0, stream>>>(x, src, dst, etype, cnt, wrel, out);
    fb_relu<<<(kNodes * kDOut + 255) / 256, 256, 0, stream>>>(out);
  }
}